// PoolModule_33397665694037
// MI455X (gfx1250) — compile-verified
//
#include <hip/hip_runtime.h>
#include <hip/hip_bf16.h>

#define BB    4
#define NP    16384
#define MC    4096          // centers per cloud = 0.25 * NP
#define R2    0.0625f       // R*R, R=0.25
#define MAXNB 64

typedef __attribute__((ext_vector_type(2))) float v2f;
typedef __attribute__((ext_vector_type(8))) float v8f;

// ---------------- K0: per-point squared norm of pos6 = [x, pos] ----------------
__global__ void pp_kernel(const float* __restrict__ x, const float* __restrict__ pos,
                          float* __restrict__ pp) {
    int i = blockIdx.x * blockDim.x + threadIdx.x;
    if (i >= BB * NP) return;
    float a0 = x[i*3+0], a1 = x[i*3+1], a2 = x[i*3+2];
    float p0 = pos[i*3+0], p1 = pos[i*3+1], p2 = pos[i*3+2];
    pp[i] = a0*a0 + a1*a1 + a2*a2 + p0*p0 + p1*p1 + p2*p2;
}

// ---------------- K1: farthest point sampling (6-D), one block per cloud -------
__global__ __launch_bounds__(512) void fps_kernel(const float* __restrict__ x,
                                                  const float* __restrict__ pos,
                                                  int* __restrict__ idx) {
    __shared__ float sv[512];
    __shared__ int   si[512];
    const int b    = blockIdx.x;
    const int tid  = threadIdx.x;
    const int base = b * NP;

    float mind[32];
#pragma unroll
    for (int j = 0; j < 32; ++j) mind[j] = 1e30f;   // distances bounded by 6 in unit cube

    int last = 0;
    for (int it = 0; it < MC; ++it) {
        if (tid == 0) idx[b * MC + it] = last;      // emit pre-update pick (matches scan)
        const int c = base + last;
        float c0 = x[c*3+0],  c1 = x[c*3+1],  c2 = x[c*3+2];
        float c3 = pos[c*3+0], c4 = pos[c*3+1], c5 = pos[c*3+2];

        float bv = -1.0f; int bi = 0;
#pragma unroll
        for (int j = 0; j < 32; ++j) {
            int i = j * 512 + tid;                  // coalesced across lanes
            int g = base + i;
            float d0 = x[g*3+0] - c0, d1 = x[g*3+1] - c1, d2 = x[g*3+2] - c2;
            float d3 = pos[g*3+0] - c3, d4 = pos[g*3+1] - c4, d5 = pos[g*3+2] - c5;
            float d  = d0*d0 + d1*d1 + d2*d2 + d3*d3 + d4*d4 + d5*d5;
            float mn = fminf(mind[j], d);
            mind[j]  = mn;
            if (mn > bv) { bv = mn; bi = i; }       // j ascending -> first max kept
        }
        sv[tid] = bv; si[tid] = bi;
        __syncthreads();
        for (int s = 256; s > 0; s >>= 1) {         // argmax, first-occurrence tie-break
            if (tid < s) {
                float ov = sv[tid + s]; int oi = si[tid + s];
                if (ov > sv[tid] || (ov == sv[tid] && oi < si[tid])) { sv[tid] = ov; si[tid] = oi; }
            }
            __syncthreads();
        }
        last = si[0];
        __syncthreads();
    }
}

// ---------------- K2: gather pos / batch outputs for the sampled centers -------
__global__ void gather_kernel(const float* __restrict__ pos, const int* __restrict__ idx,
                              float* __restrict__ out) {
    int i = blockIdx.x * blockDim.x + threadIdx.x;
    if (i >= BB * MC) return;
    int b  = i / MC;
    int gp = b * NP + idx[i];
    out[BB*MC*3 + i*3 + 0] = pos[gp*3+0];
    out[BB*MC*3 + i*3 + 1] = pos[gp*3+1];
    out[BB*MC*3 + i*3 + 2] = pos[gp*3+2];
    out[BB*MC*6 + i]       = (float)b;
}

// pos6[k] accessor: k<3 -> features x, k<6 -> coords pos, else zero pad (K=6,7)
__device__ __forceinline__ float fetch6(const float* x, const float* pos, int pt, int k) {
    if (k < 3) return x[pt*3 + k];
    if (k < 6) return pos[pt*3 + (k - 3)];
    return 0.0f;
}

// ---------------- K3: radius-neighbor masked mean via WMMA ---------------------
// One wave per 16-center tile. G = C·P^T via two v_wmma_f32_16x16x4_f32 (K=6 pad 8).
// d2 <= r2  <=>  pp[N] - 2G <= r2 - cc[M]; first-64-in-index-order via ballot rank.
__global__ __launch_bounds__(32) void pool_kernel(const float* __restrict__ x,
                                                  const float* __restrict__ pos,
                                                  const float* __restrict__ pp,
                                                  const int* __restrict__ idx,
                                                  float* __restrict__ out) {
    __shared__ float ccs[16];
    __shared__ float red[32][32];

    const int blk  = blockIdx.x;
    const int b    = blk / (MC / 16);
    const int t    = blk % (MC / 16);
    const int lane = threadIdx.x;       // 0..31
    const int row  = lane & 15;
    const int hi   = lane >> 4;         // half-wave: 0 or 1

    // --- A matrix (centers), ISA 16x4 f32 layout: lanes0-15 K=0,1 ; lanes16-31 K=2,3
    const int gC = b * NP + idx[b * MC + t * 16 + row];
    float c6[6];
    c6[0] = x[gC*3+0]; c6[1] = x[gC*3+1]; c6[2] = x[gC*3+2];
    c6[3] = pos[gC*3+0]; c6[4] = pos[gC*3+1]; c6[5] = pos[gC*3+2];
    float cc = c6[0]*c6[0]+c6[1]*c6[1]+c6[2]*c6[2]+c6[3]*c6[3]+c6[4]*c6[4]+c6[5]*c6[5];
    if (hi == 0) ccs[row] = cc;
    __syncthreads();

    const int ka = hi ? 2 : 0;
    v2f a0 = { c6[ka], c6[ka+1] };                  // K-chunk 0..3
    v2f a1 = { 0.0f, 0.0f };                        // K-chunk 4..7 (6,7 = zero pad)
    if (!hi) { a1.x = c6[4]; a1.y = c6[5]; }

    float thr[8];
#pragma unroll
    for (int r = 0; r < 8; ++r) thr[r] = R2 - ccs[r + hi * 8];

    float s0[8], s1[8], s2[8];
    int tk[8];
#pragma unroll
    for (int r = 0; r < 8; ++r) { s0[r] = 0.f; s1[r] = 0.f; s2[r] = 0.f; tk[r] = 0; }

    for (int tn = 0; tn < NP / 16; ++tn) {          // scan columns in index order
        const int pcol = tn * 16 + row;
        const int gp   = b * NP + pcol;
        float f0 = x[gp*3+0], f1 = x[gp*3+1], f2 = x[gp*3+2];
        float ppn = pp[gp];

        // B matrix (points), mirrored layout: lanes0-15 K=0,1 ; lanes16-31 K=2,3
        const int kb = hi ? 2 : 0;
        v2f b0 = { fetch6(x, pos, gp, kb), fetch6(x, pos, gp, kb + 1) };
        v2f b1 = { 0.0f, 0.0f };
        if (!hi) { b1.x = pos[gp*3+1]; b1.y = pos[gp*3+2]; }   // K=4,5

        v8f g = {};
        g = __builtin_amdgcn_wmma_f32_16x16x4_f32(false, a0, false, b0,
                                                  (short)0, g, false, false);
        g = __builtin_amdgcn_wmma_f32_16x16x4_f32(false, a1, false, b1,
                                                  (short)0, g, false, false);

        if (tn + 1 < NP / 16)                        // next point-tile prefetch
            __builtin_prefetch(&x[(gp + 16) * 3], 0, 1);

#pragma unroll
        for (int r = 0; r < 8; ++r) {                // rows M=r (lo half) / M=8+r (hi half)
            float d   = ppn - 2.0f * g[r];
            bool pred = d <= thr[r];
            unsigned bal  = __builtin_amdgcn_ballot_w32(pred);
            unsigned hm   = hi ? (bal >> 16) : (bal & 0xffffu);
            unsigned rank = __popc(hm & ((1u << row) - 1u));
            int cnt = __popc(hm);
            int rem = MAXNB - tk[r];
            if (pred && (int)rank < rem) { s0[r] += f0; s1[r] += f1; s2[r] += f2; }
            int add = cnt < rem ? cnt : rem; if (add < 0) add = 0;
            tk[r] += add;                            // uniform per half-wave
        }
    }

    // --- cross-lane reduction of per-row sums through LDS ---
#pragma unroll
    for (int r = 0; r < 8; ++r) {
        red[lane][r*4+0] = s0[r];
        red[lane][r*4+1] = s1[r];
        red[lane][r*4+2] = s2[r];
        red[lane][r*4+3] = (float)tk[r];
    }
    __syncthreads();

    if (lane < 16) {
        int M  = lane;
        int sb = (M < 8) ? 0 : 16;
        int rr = M & 7;
        float t0 = 0.f, t1 = 0.f, t2 = 0.f;
        for (int s = 0; s < 16; ++s) {
            t0 += red[sb + s][rr*4+0];
            t1 += red[sb + s][rr*4+1];
            t2 += red[sb + s][rr*4+2];
        }
        float w = red[sb][rr*4+3];                   // >=1 (center is its own neighbor)
        int oi = b * MC + t * 16 + M;
        out[oi*3+0] = t0 / w;
        out[oi*3+1] = t1 / w;
        out[oi*3+2] = t2 / w;
    }
}

extern "C" void kernel_launch(void* const* d_in, const int* in_sizes, int n_in,
                              void* d_out, int out_size, void* d_ws, size_t ws_size,
                              hipStream_t stream) {
    const float* x   = (const float*)d_in[0];
    const float* pos = (const float*)d_in[1];
    // d_in[2] (batch) is implied by the fixed equal-size layout; unused.
    float* out = (float*)d_out;

    float* pp  = (float*)d_ws;                                   // BB*NP floats
    int*   idx = (int*)((char*)d_ws + (size_t)BB * NP * sizeof(float)); // BB*MC ints

    pp_kernel    <<<(BB * NP + 255) / 256, 256, 0, stream>>>(x, pos, pp);
    fps_kernel   <<<BB, 512, 0, stream>>>(x, pos, idx);
    gather_kernel<<<(BB * MC + 255) / 256, 256, 0, stream>>>(pos, idx, out);
    pool_kernel  <<<BB * (MC / 16), 32, 0, stream>>>(x, pos, pp, idx, out);
}